// GraphAttentionLayer_36335423324410
// MI455X (gfx1250) — compile-verified
//
#include <hip/hip_runtime.h>

typedef __attribute__((ext_vector_type(2))) float v2f;
typedef __attribute__((ext_vector_type(8))) float v8f;

static constexpr float kAlpha = 0.2f;
static constexpr float kNeg   = -9.0e15f;
static constexpr float kEps   = 1e-5f;
static constexpr int   kInF   = 256;
static constexpr int   kOutF  = 64;

#define CH        64            // j-chunk staged in LDS per block
#define LDSTRIDE  72            // padded row stride (floats): 2*72 % 64 == 16
                                // -> lanes 0-15 / 16-31 hit disjoint bank groups

// ---- async global->LDS path (guarded; falls back to sync staging) ----------
#if defined(__AMDGCN__) && __has_builtin(__builtin_amdgcn_global_load_async_to_lds_b128)
#define ASYNC_COPY 1
#else
#define ASYNC_COPY 0
#endif

#if ASYNC_COPY
typedef __attribute__((ext_vector_type(4))) int v4i;
typedef __attribute__((address_space(1))) v4i* gp_v4i;   // global int4*
typedef __attribute__((address_space(3))) v4i* lp_v4i;   // LDS int4*
#if __has_builtin(__builtin_amdgcn_s_wait_asynccnt)
#define WAIT_ASYNC(n) __builtin_amdgcn_s_wait_asynccnt(n)
#else
#define WAIT_ASYNC(n) asm volatile("s_wait_asynccnt %0" ::"n"(n))
#endif
#else
#define WAIT_ASYNC(n)
#endif

__device__ __forceinline__ float lrelu(float e) { return fmaxf(e, kAlpha * e); }

// ---------------------------------------------------------------------------
// Kernel A: h = input @ W  (fp32 WMMA 16x16x4), then f = h@a1, g = h@a2.
// One wave computes a 16x64 tile of h. 4 waves per block.
// ---------------------------------------------------------------------------
__global__ __launch_bounds__(128) void gat_h_proj(
    const float* __restrict__ input, const float* __restrict__ W,
    const float* __restrict__ a1, const float* __restrict__ a2,
    float* __restrict__ h, float* __restrict__ fv, float* __restrict__ gv, int n)
{
  const int lane = threadIdx.x & 31;
  const int wv   = threadIdx.x >> 5;
  const int ib   = blockIdx.x * 4 + wv;     // 16-row tile index
  if (ib * 16 >= n) return;                 // wave-uniform: EXEC stays all-ones
  const int r16 = lane & 15;
  const int hi  = lane >> 4;

  const float* Arow = input + (size_t)(ib * 16 + r16) * kInF;

  v8f acc[4];
#pragma unroll
  for (int t = 0; t < 4; ++t) acc[t] = 0.0f;

#pragma unroll 4
  for (int k0 = 0; k0 < kInF; k0 += 4) {
    // A 16x4 tile: lane holds K = k0 + 2*hi + {0,1} for row r16 (one b64 load)
    v2f a = *(const v2f*)(Arow + (k0 + 2 * hi));
    const float* w0 = W + (size_t)(k0 + 2 * hi) * kOutF + r16;
#pragma unroll
    for (int t = 0; t < 4; ++t) {
      v2f b;                                // B 4x16 tile, cols 16t..16t+15
      b.x = w0[t * 16];                     // K = k0 + 2*hi
      b.y = w0[t * 16 + kOutF];             // K = k0 + 2*hi + 1
      acc[t] = __builtin_amdgcn_wmma_f32_16x16x4_f32(
          false, a, false, b, (short)0, acc[t], false, false);
    }
  }

  float a1c[4], a2c[4];
#pragma unroll
  for (int t = 0; t < 4; ++t) { a1c[t] = a1[t * 16 + r16]; a2c[t] = a2[t * 16 + r16]; }

#pragma unroll
  for (int v = 0; v < 8; ++v) {
    const int mrow = ib * 16 + v + 8 * hi;  // C-layout row for this lane/vgpr
    float pf = 0.f, pg = 0.f;
#pragma unroll
    for (int t = 0; t < 4; ++t) {
      float hval = acc[t][v];
      pf += hval * a1c[t];
      pg += hval * a2c[t];
      h[(size_t)mrow * kOutF + t * 16 + r16] = hval;
    }
    // reduce over the 16 lanes that share this row (columns 0..63)
#pragma unroll
    for (int mk = 1; mk < 16; mk <<= 1) {
      pf += __shfl_xor(pf, mk, 16);
      pg += __shfl_xor(pg, mk, 16);
    }
    if (r16 == 0) { fv[mrow] = pf; gv[mrow] = pg; }
  }
}

// ---------------------------------------------------------------------------
// Kernel B1: per-row masked-softmax statistics: m_i = max_j e_ij,
// s_i = sum_j exp(e_ij - m_i), with e_ij = lrelu(f_i+g_j) or NEG if !adj.
// One wave per row, online (branchless) max/sum, int4/float4 loads.
// ---------------------------------------------------------------------------
__global__ __launch_bounds__(256) void gat_row_max_sum(
    const int* __restrict__ adj, const float* __restrict__ fv,
    const float* __restrict__ gv, float* __restrict__ mrow,
    float* __restrict__ srow, int n)
{
  const int lane = threadIdx.x & 31;
  const int row  = blockIdx.x * (blockDim.x >> 5) + (threadIdx.x >> 5);
  if (row >= n) return;
  const float fi = fv[row];
  const int* arow = adj + (size_t)row * n;

  float m = -3.0e38f, s = 0.0f;
  for (int k = lane * 4; k < n; k += 128) {
    int4   av = *(const int4*)(arow + k);
    float4 gq = *(const float4*)(gv + k);
    float e0 = lrelu(fi + gq.x); e0 = av.x > 0 ? e0 : kNeg;
    float e1 = lrelu(fi + gq.y); e1 = av.y > 0 ? e1 : kNeg;
    float e2 = lrelu(fi + gq.z); e2 = av.z > 0 ? e2 : kNeg;
    float e3 = lrelu(fi + gq.w); e3 = av.w > 0 ? e3 : kNeg;
    float cm = fmaxf(fmaxf(e0, e1), fmaxf(e2, e3));
    float nm = fmaxf(m, cm);
    s = s * __expf(m - nm) + __expf(e0 - nm) + __expf(e1 - nm)
        + __expf(e2 - nm) + __expf(e3 - nm);
    m = nm;
  }
#pragma unroll
  for (int mk = 16; mk >= 1; mk >>= 1) {
    float mo = __shfl_xor(m, mk, 32);
    float so = __shfl_xor(s, mk, 32);
    float nm = fmaxf(m, mo);
    s = s * __expf(m - nm) + so * __expf(mo - nm);
    m = nm;
  }
  if (lane == 0) { mrow[row] = m; srow[row] = s; }
}

// ---------------------------------------------------------------------------
// Stage one 64-row chunk of h (64 floats/row) into LDS with padded stride.
// 128 threads x 8 b128 units. Async (ASYNCcnt) when available.
// ---------------------------------------------------------------------------
__device__ __forceinline__ void stage_chunk(const float* __restrict__ hsrc,
                                            float* dstbuf, int tid)
{
#pragma unroll
  for (int i = 0; i < 8; ++i) {
    const int u   = tid + i * 128;          // b128 unit index 0..1023
    const int row = u >> 4;
    const int c4  = (u & 15) * 4;
    const float* src = hsrc + row * kOutF + c4;
    float*       dst = dstbuf + row * LDSTRIDE + c4;
#if ASYNC_COPY
    __builtin_amdgcn_global_load_async_to_lds_b128(
        (gp_v4i)(void*)src, (lp_v4i)(void*)dst, 0, 0);
#else
    *(float4*)dst = *(const float4*)src;
#endif
  }
}

// ---------------------------------------------------------------------------
// Kernel B2: h' = softmax(e) @ h fused with LayerNorm + ELU.
// One wave owns 16 output rows; 4 waves/block share double-buffered LDS
// staging of h chunks (async global->LDS). Attention probs are generated
// directly in the WMMA A-operand layout; B operands come from LDS.
// ---------------------------------------------------------------------------
__global__ __launch_bounds__(128) void gat_attn_norm(
    const int* __restrict__ adj, const float* __restrict__ h,
    const float* __restrict__ fv, const float* __restrict__ gv,
    const float* __restrict__ mrow, const float* __restrict__ srow,
    const float* __restrict__ gamma, const float* __restrict__ beta,
    float* __restrict__ out, int n)
{
  __shared__ float hbuf[2][CH * LDSTRIDE];  // 2 x 18KB

  const int tid  = threadIdx.x;
  const int lane = tid & 31;
  const int wv   = tid >> 5;
  const int ib   = blockIdx.x * 4 + wv;
  const bool valid = (ib * 16 < n);         // wave-uniform predicate
  const int r16 = lane & 15;
  const int hi  = lane >> 4;

  const int   arowi = ib * 16 + r16;        // A-layout row for this lane
  const float fi = valid ? fv[arowi] : 0.f;
  const float mi = valid ? mrow[arowi] : 0.f;
  const int* arow = adj + (size_t)(valid ? arowi : 0) * n;

  v8f acc[4];
#pragma unroll
  for (int t = 0; t < 4; ++t) acc[t] = 0.0f;

  const int nc = n / CH;
  stage_chunk(h, hbuf[0], tid);             // prologue: chunk 0

  for (int c = 0; c < nc; ++c) {
    const float* cur = hbuf[c & 1];
    if (c + 1 < nc) {
      stage_chunk(h + (size_t)(c + 1) * CH * kOutF, hbuf[(c + 1) & 1], tid);
      WAIT_ASYNC(8);                        // chunk c's 8 requests are done
    } else {
      WAIT_ASYNC(0);
    }
    __syncthreads();                        // chunk c visible to all waves

    const int j0 = c * CH;
    if (valid) {
      if (j0 + 2048 < n) __builtin_prefetch(arow + j0 + 2048, 0, 1);
#pragma unroll 4
      for (int js = 0; js < CH; js += 4) {
        const int jj = js + 2 * hi;         // in-chunk K pair for this lane
        int2 av = *(const int2*)(arow + j0 + jj);
        v2f  gq = *(const v2f*)(gv + j0 + jj);
        float e0 = lrelu(fi + gq.x); e0 = av.x > 0 ? e0 : kNeg;
        float e1 = lrelu(fi + gq.y); e1 = av.y > 0 ? e1 : kNeg;
        v2f a;                              // unnormalized softmax probs
        a.x = __expf(e0 - mi);
        a.y = __expf(e1 - mi);
        const float* hr = cur + jj * LDSTRIDE + r16;
#pragma unroll
        for (int t = 0; t < 4; ++t) {
          v2f b;                            // B rows jj, jj+1 from LDS
          b.x = hr[t * 16];
          b.y = hr[t * 16 + LDSTRIDE];
          acc[t] = __builtin_amdgcn_wmma_f32_16x16x4_f32(
              false, a, false, b, (short)0, acc[t], false, false);
        }
      }
    }
    __syncthreads();                        // safe to overwrite buf (c+2 issue)
  }

  if (!valid) return;

  float gam[4], bet[4];
#pragma unroll
  for (int t = 0; t < 4; ++t) { gam[t] = gamma[t * 16 + r16]; bet[t] = beta[t * 16 + r16]; }

#pragma unroll
  for (int v = 0; v < 8; ++v) {
    const int orow = ib * 16 + v + 8 * hi;  // C-layout row
    const float inv = 1.0f / srow[orow];    // softmax normalization
    float x[4];
    float sum = 0.f;
#pragma unroll
    for (int t = 0; t < 4; ++t) { x[t] = acc[t][v] * inv; sum += x[t]; }
#pragma unroll
    for (int mk = 1; mk < 16; mk <<= 1) sum += __shfl_xor(sum, mk, 16);
    const float mean = sum * (1.0f / kOutF);
    float ss = 0.f;
#pragma unroll
    for (int t = 0; t < 4; ++t) { float d = x[t] - mean; ss += d * d; }
#pragma unroll
    for (int mk = 1; mk < 16; mk <<= 1) ss += __shfl_xor(ss, mk, 16);
    const float rs = rsqrtf(ss * (1.0f / kOutF) + kEps);
#pragma unroll
    for (int t = 0; t < 4; ++t) {
      float y = (x[t] - mean) * rs * gam[t] + bet[t];
      y = y > 0.f ? y : expm1f(y);          // ELU (alpha=1)
      out[(size_t)orow * kOutF + t * 16 + r16] = y;
    }
  }
}

// ---------------------------------------------------------------------------
extern "C" void kernel_launch(void* const* d_in, const int* in_sizes, int n_in,
                              void* d_out, int out_size, void* d_ws, size_t ws_size,
                              hipStream_t stream) {
  const float* input = (const float*)d_in[0];
  const int*   adj   = (const int*)d_in[1];
  const float* W     = (const float*)d_in[2];
  const float* a1    = (const float*)d_in[3];
  const float* a2    = (const float*)d_in[4];
  const float* gamma = (const float*)d_in[5];
  const float* beta  = (const float*)d_in[6];
  float* out = (float*)d_out;

  const int n = in_sizes[0] / kInF;         // 8192

  // workspace partition: h[N,64], f[N], g[N], m[N], s[N]  (~2.13 MB total)
  float* h  = (float*)d_ws;
  float* fv = h + (size_t)n * kOutF;
  float* gv = fv + n;
  float* mr = gv + n;
  float* sr = mr + n;

  const int nb16 = n / 16;                  // 512 row tiles
  dim3 blk(128);                            // 4 waves / block
  dim3 grdA((nb16 + 3) / 4);

  gat_h_proj<<<grdA, blk, 0, stream>>>(input, W, a1, a2, h, fv, gv, n);
  gat_row_max_sum<<<dim3((n + 7) / 8), dim3(256), 0, stream>>>(adj, fv, gv, mr, sr, n);
  gat_attn_norm<<<grdA, blk, 0, stream>>>(adj, h, fv, gv, mr, sr, gamma, beta, out, n);
}